// UNetBlockIntegrated_87677462381216
// MI455X (gfx1250) — compile-verified
//
#include <hip/hip_runtime.h>
#include <hip/hip_bf16.h>

// ---------------------------------------------------------------------------
// Problem constants (from reference)
// B=16, C=256, H=W=32, E=512, GN_GROUPS=32, N_QUBITS=6 (DIM=64),
// N_DATA=4, N_LAYERS=4, N_GROUPS=8  -> ang channels = 32, meas channels = 32
// ---------------------------------------------------------------------------

typedef __attribute__((ext_vector_type(2))) float v2f;
typedef __attribute__((ext_vector_type(8))) float v8f;

// ---- compile-time CNOT-ring permutation, folded into register relabeling ----
constexpr int fperm(int j) {
    int f = j;
    for (int c = 0; c < 6; ++c) {
        int t = (c + 1) % 6;
        int cbit = (f >> (5 - c)) & 1;
        if (cbit) f ^= (1 << (5 - t));
    }
    return f;
}

struct SigT { int s[5][64]; };  // sigma for layers 0..3 + final (index 4)

constexpr SigT make_sig() {
    SigT S{};
    int cur[64] = {};
    for (int i = 0; i < 64; ++i) cur[i] = i;
    for (int l = 0; l < 4; ++l) {
        for (int i = 0; i < 64; ++i) S.s[l][i] = cur[i];
        int nxt[64] = {};
        // new[f(j)] = old[j]  =>  sigma'(f(j)) = sigma(j)
        for (int j = 0; j < 64; ++j) nxt[fperm(j)] = cur[j];
        for (int i = 0; i < 64; ++i) cur[i] = nxt[i];
    }
    for (int i = 0; i < 64; ++i) S.s[4][i] = cur[i];
    return S;
}

constexpr SigT SIG = make_sig();

static __device__ __forceinline__ v8f wmma_f32x4(v2f a, v2f b, v8f c) {
    // D = A(16x4 f32) * B(4x16 f32) + C(16x16 f32)
    return __builtin_amdgcn_wmma_f32_16x16x4_f32(false, a, false, b,
                                                 (short)0, c, false, false);
}

// ---------------------------------------------------------------------------
// K0: GroupNorm statistics. One block per (b, gn_group); contiguous 8192 floats
// ---------------------------------------------------------------------------
__global__ __launch_bounds__(256) void qk_stats(const float* __restrict__ x,
                                                float* __restrict__ stats) {
    const int bg = blockIdx.x;                 // b*32 + gn
    const float* base = x + (size_t)bg * 8192; // 8 channels * 1024 pixels
    float s = 0.f, sq = 0.f;
    for (int i = threadIdx.x; i < 8192; i += 256) {
        float v = base[i];
        s += v; sq += v * v;
    }
    __shared__ float sh[256], sh2[256];
    sh[threadIdx.x] = s; sh2[threadIdx.x] = sq;
    __syncthreads();
    for (int off = 128; off > 0; off >>= 1) {
        if (threadIdx.x < off) {
            sh[threadIdx.x]  += sh[threadIdx.x + off];
            sh2[threadIdx.x] += sh2[threadIdx.x + off];
        }
        __syncthreads();
    }
    if (threadIdx.x == 0) {
        float m   = sh[0] * (1.f / 8192.f);
        float var = sh2[0] * (1.f / 8192.f) - m * m;
        stats[bg * 2 + 0] = m;
        stats[bg * 2 + 1] = rsqrtf(var + 1e-5f);
    }
}

// ---------------------------------------------------------------------------
// K1: style = SiLU(emb) @ w_style^T + b_style ; bh = 0.5*(theta + style)
// One block per batch. bh layout: [b][g][l][q] (o = g*24 + l*6 + q)
// ---------------------------------------------------------------------------
__global__ __launch_bounds__(256) void qk_style(const float* __restrict__ emb,
                                                const float* __restrict__ w_style,
                                                const float* __restrict__ b_style,
                                                const float* __restrict__ theta,
                                                float* __restrict__ bh) {
    const int b = blockIdx.x;
    __shared__ float se[512];
    for (int e = threadIdx.x; e < 512; e += blockDim.x) {
        float v = emb[b * 512 + e];
        se[e] = v / (1.f + __expf(-v));        // SiLU
    }
    __syncthreads();
    const int o = threadIdx.x;
    if (o < 192) {
        float acc = b_style[o];
        const float* w = w_style + (size_t)o * 512;
        for (int e = 0; e < 512; ++e) acc = fmaf(se[e], w[e], acc);
        bh[b * 192 + o] = 0.5f * (theta[o] + acc);
    }
}

// ---------------------------------------------------------------------------
// K2: fused  norm -> ang GEMM (WMMA) -> quantum circuit -> meas
//            -> out GEMM (WMMA) + bias + residual
// Grid: (32 pixel-tiles, 16 batches), 256 threads = 8 wave32
// ---------------------------------------------------------------------------
__global__ __launch_bounds__(256) void qk_fused(const float* __restrict__ x,
                                                const float* __restrict__ gamma,
                                                const float* __restrict__ beta,
                                                const float* __restrict__ w_in,
                                                const float* __restrict__ b_in,
                                                const float* __restrict__ w_out,
                                                const float* __restrict__ b_out,
                                                const float* __restrict__ stats,
                                                const float* __restrict__ bh,
                                                float* __restrict__ out) {
    __shared__ float h[256][33];   // normalized activations [channel][pixel]
    __shared__ float am[32][33];   // ang, then reused for meas [chan32][pixel]
    __shared__ float bhs[192];     // half-angle bases for this batch

    const int b    = blockIdx.y;
    const int pix0 = blockIdx.x * 32;
    const int tid  = threadIdx.x;
    const int lane = tid & 31;
    const int wave = tid >> 5;
    const int half = lane >> 4;    // WMMA lane half (K offset)
    const int lid  = lane & 15;    // WMMA lane within half (M/N index)

    if (tid < 192) bhs[tid] = bh[b * 192 + tid];

    // ---- load + GroupNorm: h[c][p] ----
    {
        const int p = lane;        // 32 consecutive pixels per wave row
        for (int c = wave; c < 256; c += 8) {
            const int gn = c >> 3;
            const float mean = stats[(b * 32 + gn) * 2 + 0];
            const float rs   = stats[(b * 32 + gn) * 2 + 1];
            const size_t gi  = ((size_t)b * 256 + c) * 1024 + (size_t)(pix0 + p);
            h[c][p] = (x[gi] - mean) * rs * gamma[c] + beta[c];
        }
    }
    __syncthreads();

    // ---- ang GEMM: [32 out] x [K=256] x [32 pixels], WMMA f32 16x16x4 ----
    // wave -> (Mtile, Ntile, Khalf)
    {
        const int mt = wave & 1, nt = (wave >> 1) & 1, kh = wave >> 2;
        const int kbase = kh * 128;
        const int arow = mt * 16 + lid;
        const int bcol = nt * 16 + lid;
        v8f acc = {};
#pragma unroll 8
        for (int ks = 0; ks < 128; ks += 4) {
            const int k = kbase + ks + 2 * half;
            v2f A; A.x = w_in[arow * 256 + k]; A.y = w_in[arow * 256 + k + 1];
            v2f Bv; Bv.x = h[k][bcol];         Bv.y = h[k + 1][bcol];
            acc = wmma_f32x4(A, Bv, acc);
        }
        if (kh == 0) {
#pragma unroll
            for (int r = 0; r < 8; ++r) {
                const int o = mt * 16 + r + 8 * half;
                am[o][bcol] = acc[r] + b_in[o];
            }
        }
        __syncthreads();
        if (kh == 1) {
#pragma unroll
            for (int r = 0; r < 8; ++r) {
                const int o = mt * 16 + r + 8 * half;
                am[o][bcol] += acc[r];
            }
        }
        __syncthreads();
    }

    // ---- quantum circuit: one (pixel, group) per thread, state in VGPRs ----
    const int g = tid & 7;         // group
    const int p = tid >> 3;        // pixel within tile
    float angp[4];
#pragma unroll
    for (int d = 0; d < 4; ++d) angp[d] = am[g * 4 + d][p];
    __syncthreads();               // all reads of `am` done before meas writes

    float st[64];
#pragma unroll
    for (int i = 0; i < 64; ++i) st[i] = 0.f;
    st[0] = 1.f;

#pragma unroll
    for (int l = 0; l < 4; ++l) {
#pragma unroll
        for (int q = 0; q < 6; ++q) {
            float ha = bhs[g * 24 + l * 6 + q];
            if (q < 4) ha += 0.5f * angp[q];
            const float cs = __cosf(ha);
            const float sn = __sinf(ha);
            const int stride = 1 << (5 - q);
#pragma unroll
            for (int i = 0; i < 64; ++i) {
                if ((i & stride) == 0) {
                    const int r0 = SIG.s[l][i];
                    const int r1 = SIG.s[l][i + stride];
                    const float a0 = st[r0], a1 = st[r1];
                    st[r0] = cs * a0 - sn * a1;
                    st[r1] = sn * a0 + cs * a1;
                }
            }
            // CNOT-ring permutation between layers is folded into SIG (free)
        }
    }

    // ---- measurement: meas[d] = sum_i zsign(i,d) * |st_i|^2 ----
    float m[4] = {0.f, 0.f, 0.f, 0.f};
#pragma unroll
    for (int i = 0; i < 64; ++i) {
        const float v = st[SIG.s[4][i]];
        const float pr = v * v;
#pragma unroll
        for (int d = 0; d < 4; ++d) {
            if ((i >> (5 - d)) & 1) m[d] -= pr; else m[d] += pr;
        }
    }
#pragma unroll
    for (int d = 0; d < 4; ++d) am[g * 4 + d][p] = m[d];
    __syncthreads();

    // ---- out GEMM: [256 out] x [K=32] x [32 pixels], + b_out + residual ----
    // 32 tile-jobs (16 Mtiles x 2 Ntiles), 4 per wave
#pragma unroll
    for (int i = 0; i < 4; ++i) {
        const int jj = wave + 8 * i;
        const int mt = jj & 15;
        const int nt = jj >> 4;
        const int arow = mt * 16 + lid;
        const int bcol = nt * 16 + lid;
        v8f acc = {};
#pragma unroll
        for (int k = 0; k < 32; k += 4) {
            const int kk = k + 2 * half;
            v2f A; A.x = w_out[arow * 32 + kk]; A.y = w_out[arow * 32 + kk + 1];
            v2f Bv; Bv.x = am[kk][bcol];        Bv.y = am[kk + 1][bcol];
            acc = wmma_f32x4(A, Bv, acc);
        }
#pragma unroll
        for (int r = 0; r < 8; ++r) {
            const int o = mt * 16 + r + 8 * half;
            const size_t gi = ((size_t)b * 256 + o) * 1024 + (size_t)(pix0 + bcol);
            out[gi] = acc[r] + b_out[o] + x[gi];
        }
    }
}

// ---------------------------------------------------------------------------
extern "C" void kernel_launch(void* const* d_in, const int* in_sizes, int n_in,
                              void* d_out, int out_size, void* d_ws, size_t ws_size,
                              hipStream_t stream) {
    const float* x       = (const float*)d_in[0];
    const float* emb     = (const float*)d_in[1];
    const float* gamma   = (const float*)d_in[2];
    const float* beta    = (const float*)d_in[3];
    const float* w_in    = (const float*)d_in[4];
    const float* b_in    = (const float*)d_in[5];
    const float* theta   = (const float*)d_in[6];
    const float* w_style = (const float*)d_in[7];
    const float* b_style = (const float*)d_in[8];
    const float* w_out   = (const float*)d_in[9];
    const float* b_out   = (const float*)d_in[10];
    float* out = (float*)d_out;

    float* stats = (float*)d_ws;        // 512 * 2 floats
    float* bh    = stats + 1024;        // 16 * 192 floats

    qk_stats<<<dim3(512), dim3(256), 0, stream>>>(x, stats);
    qk_style<<<dim3(16), dim3(256), 0, stream>>>(emb, w_style, b_style, theta, bh);
    qk_fused<<<dim3(32, 16), dim3(256), 0, stream>>>(x, gamma, beta, w_in, b_in,
                                                     w_out, b_out, stats, bh, out);
}